// LightGCL_Encoder_35003983462563
// MI455X (gfx1250) — compile-verified
//
#include <hip/hip_runtime.h>
#include <hip/hip_fp16.h>

// LightGCL encoder for MI455X (gfx1250, wave32).
// - SpMM scatter: edge-parallel, float4-coalesced gathers (tables are L2-resident:
//   12.8/25.6 MB << 192 MB L2), scatter via HW global_atomic_add_f32 (no return).
// - Low-rank propagation: V_WMMA_F32_16X16X4_F32 (fp32 matrix core), K padded 5->8,
//   fused with the E_sum += Z accumulation pass (same [N,64] rows).

typedef __attribute__((ext_vector_type(8)))  float    v8f;
typedef __attribute__((ext_vector_type(2)))  float    v2f;
typedef __attribute__((ext_vector_type(16))) _Float16 v16h;

#define DROP_RATE 0.2f
#define DDIM 64

__device__ __forceinline__ void atomAddF32(float* p, float v)
{
    // Always lowers to the hardware fp32 atomic (global_atomic_add_f32),
    // never a CAS loop.
    unsafeAtomicAdd(p, v);
}

// ---------------------------------------------------------------------------
// SpMM with sparse dropout: dst[dst_idx[e]] += (mask[e] > p ? vals[e] : 0) * src[src_idx[e]]
// 16 lanes per edge, one float4 (16B) per lane -> one 256B row per 16 lanes,
// perfectly coalesced. Edge scalars are same-address loads within the lane
// group -> single broadcast request per wave.
// ---------------------------------------------------------------------------
__global__ void spmm_scatter(const float* __restrict__ src,      // [Nsrc, 64]
                             float*       __restrict__ dst,      // [Ndst, 64] (zeroed)
                             const int*   __restrict__ dst_idx,  // [E]
                             const int*   __restrict__ src_idx,  // [E]
                             const float* __restrict__ vals,     // [E]
                             const float* __restrict__ mask,     // [E]
                             int E)
{
    long long gid = (long long)blockIdx.x * blockDim.x + threadIdx.x;
    long long e   = gid >> 4;          // 16 threads per edge
    int       j   = (int)(gid & 15);   // which float4 of the row
    if (e >= E) return;
    float m = mask[e];
    if (m <= DROP_RATE) return;        // dropped edge: no gather, no atomics
    float w = vals[e];
    int   s = src_idx[e];
    int   d = dst_idx[e];
    const float4 v = *(const float4*)(src + (long long)s * DDIM + j * 4);
    float* p = dst + (long long)d * DDIM + j * 4;
    atomAddF32(p + 0, w * v.x);
    atomAddF32(p + 1, w * v.y);
    atomAddF32(p + 2, w * v.z);
    atomAddF32(p + 3, w * v.w);
}

// ---------------------------------------------------------------------------
// Tpad[8,64] (zero-padded) = P[Q,N] @ Emb[N,64].  One thread per column d with
// Q accumulators: Emb read exactly once; P[q*N+i] is wave-uniform (broadcast).
// blockDim = 256 -> 4 row-chunks per block.
// ---------------------------------------------------------------------------
__global__ void low_rank_reduce(const float* __restrict__ P,    // [Q, N]
                                const float* __restrict__ Emb,  // [N, 64]
                                float*       __restrict__ Tpad, // [8, 64] (zeroed)
                                int N, int Q)
{
    int d      = threadIdx.x & 63;
    int sub    = threadIdx.x >> 6;                 // 0..3
    int stride = gridDim.x * 4;
    long long i0 = (long long)blockIdx.x * 4 + sub;

    float acc[8];
    #pragma unroll
    for (int q = 0; q < 8; ++q) acc[q] = 0.f;

    for (long long i = i0; i < N; i += stride) {
        float e = Emb[i * DDIM + d];
        for (int q = 0; q < Q; ++q)
            acc[q] += P[(long long)q * N + i] * e;
    }
    for (int q = 0; q < Q; ++q)
        atomAddF32(&Tpad[q * DDIM + d], acc[q]);
}

// ---------------------------------------------------------------------------
// Fused per-row update over a 16-row tile (one wave per tile):
//   Gsum[N,64] += A[N,Q] @ Tpad[8,64]     (WMMA f32 16x16x4, k=0..3 / 4..7)
//   Esum[N,64] += Z[N,64]
// A-frag (32b A, 16x4): lane L: M=L&15; VGPR0/1 hold K=(L<16 ? 0,1 : 2,3).
// C/D: VGPR v: lanes0-15 -> M=v, lanes16-31 -> M=v+8; N=lane&15.
// All fragment loads are unconditional (arithmetic masking, no exec branches).
// ---------------------------------------------------------------------------
__global__ void lowrank_wmma_fused(const float* __restrict__ A,    // [Nrows, Q]
                                   const float* __restrict__ Tpad, // [8, 64] zero-padded
                                   const float* __restrict__ Z,    // [Nrows, 64]
                                   float*       __restrict__ Esum, // [Nrows, 64] (+=Z)
                                   float*       __restrict__ Gsum, // [Nrows, 64] (+=A@T)
                                   int Nrows, int Q)
{
    int wave = blockIdx.x * (blockDim.x >> 5) + (threadIdx.x >> 5);
    int lane = threadIdx.x & 31;
    int r0   = wave * 16;
    if (r0 >= Nrows) return;           // wave-uniform exit (EXEC stays full)

    int   m       = lane & 15;         // M (A) / N (B,D) sub-index
    int   koff    = (lane >> 4) << 1;  // lanes 16-31 hold K+2,K+3
    int   row_a   = r0 + m;
    float rowmask = (row_a < Nrows) ? 1.f : 0.f;
    long long abase = (long long)(row_a < Nrows ? row_a : 0) * Q;

    v8f acc[4] = {};                   // 4 N-tiles of 16 cover D=64

    #pragma unroll
    for (int kb = 0; kb < 8; kb += 4) {
        int   k0  = kb + koff, k1 = k0 + 1;
        float m0  = (k0 < Q) ? rowmask : 0.f;
        float m1  = (k1 < Q) ? rowmask : 0.f;
        int   kk0 = (k0 < Q) ? k0 : 0;     // clamped safe indices
        int   kk1 = (k1 < Q) ? k1 : 0;
        v2f a;
        a.x = A[abase + kk0] * m0;         // unconditional loads, masked in VALU
        a.y = A[abase + kk1] * m1;
        #pragma unroll
        for (int n = 0; n < 4; ++n) {
            int nn = n * 16 + m;
            v2f b;
            b.x = Tpad[k0 * DDIM + nn];    // rows >= Q are zero-padded
            b.y = Tpad[k1 * DDIM + nn];
            acc[n] = __builtin_amdgcn_wmma_f32_16x16x4_f32(
                false, a, false, b, (short)0, acc[n], false, false);
        }
    }

    int rbase = r0 + ((lane >> 4) << 3);   // lanes 16-31 -> M += 8
    #pragma unroll
    for (int v = 0; v < 8; ++v) {
        int row = rbase + v;
        if (row < Nrows) {
            long long rb = (long long)row * DDIM;
            #pragma unroll
            for (int n = 0; n < 4; ++n) {
                int col = n * 16 + m;
                Gsum[rb + col] += acc[n][v];
                Esum[rb + col] += Z[rb + col];
            }
        }
    }
}

extern "C" void kernel_launch(void* const* d_in, const int* in_sizes, int n_in,
                              void* d_out, int out_size, void* d_ws, size_t ws_size,
                              hipStream_t stream)
{
    const float* user_emb = (const float*)d_in[0];
    const float* item_emb = (const float*)d_in[1];
    const float* vals     = (const float*)d_in[2];
    const float* u_mul_s  = (const float*)d_in[3];
    const float* v_mul_s  = (const float*)d_in[4];
    const float* ut       = (const float*)d_in[5];
    const float* vt       = (const float*)d_in[6];
    const float* dropm    = (const float*)d_in[7];
    const int*   rows     = (const int*)d_in[8];
    const int*   cols     = (const int*)d_in[9];

    const int U = in_sizes[0] / DDIM;       // 100000
    const int I = in_sizes[1] / DDIM;       // 50000
    const int E = in_sizes[2];              // 2.5M
    const int Q = in_sizes[3] / U;          // 5

    // Outputs, concatenated flat: E_u_sum, E_i_sum, G_u_sum, G_i_sum
    float* Eu_sum = (float*)d_out;
    float* Ei_sum = Eu_sum + (size_t)U * DDIM;
    float* Gu_sum = Ei_sum + (size_t)I * DDIM;
    float* Gi_sum = Gu_sum + (size_t)U * DDIM;

    // Workspace: double-buffered layer embeddings + 2 padded [8,64] T tiles
    float* ws = (float*)d_ws;
    float* Ecur_u = ws;  ws += (size_t)U * DDIM;
    float* Ecur_i = ws;  ws += (size_t)I * DDIM;
    float* Znxt_u = ws;  ws += (size_t)U * DDIM;
    float* Znxt_i = ws;  ws += (size_t)I * DDIM;
    float* T_iu   = ws;  ws += 8 * DDIM;   // vt @ E_i  (for G_u)
    float* T_ui   = ws;  ws += 8 * DDIM;   // ut @ E_u  (for G_i)

    const size_t ubytes = (size_t)U * DDIM * sizeof(float);
    const size_t ibytes = (size_t)I * DDIM * sizeof(float);

    // Initialize running sums and layer-0 embeddings (capture-safe async d2d)
    hipMemcpyAsync(Ecur_u, user_emb, ubytes, hipMemcpyDeviceToDevice, stream);
    hipMemcpyAsync(Ecur_i, item_emb, ibytes, hipMemcpyDeviceToDevice, stream);
    hipMemcpyAsync(Eu_sum, user_emb, ubytes, hipMemcpyDeviceToDevice, stream);
    hipMemcpyAsync(Ei_sum, item_emb, ibytes, hipMemcpyDeviceToDevice, stream);
    hipMemcpyAsync(Gu_sum, user_emb, ubytes, hipMemcpyDeviceToDevice, stream);
    hipMemcpyAsync(Gi_sum, item_emb, ibytes, hipMemcpyDeviceToDevice, stream);

    for (int layer = 0; layer < 2; ++layer) {
        hipMemsetAsync(Znxt_u, 0, ubytes, stream);
        hipMemsetAsync(Znxt_i, 0, ibytes, stream);
        hipMemsetAsync(T_iu, 0, 8 * DDIM * sizeof(float), stream);
        hipMemsetAsync(T_ui, 0, 8 * DDIM * sizeof(float), stream);

        // --- SpMMs (dominant cost; L2-resident gather + HW fp32 atomics) ---
        long long work = (long long)E * 16;
        int  blk  = 256;
        unsigned grid = (unsigned)((work + blk - 1) / blk);
        spmm_scatter<<<grid, blk, 0, stream>>>(Ecur_i, Znxt_u, rows, cols, vals,
                                               dropm + (size_t)(layer * 2 + 0) * E, E);
        spmm_scatter<<<grid, blk, 0, stream>>>(Ecur_u, Znxt_i, cols, rows, vals,
                                               dropm + (size_t)(layer * 2 + 1) * E, E);

        // --- rank-Q reductions into padded staging tiles ---
        low_rank_reduce<<<240, 256, 0, stream>>>(vt, Ecur_i, T_iu, I, Q);
        low_rank_reduce<<<240, 256, 0, stream>>>(ut, Ecur_u, T_ui, U, Q);

        // --- fused: Gsum += A @ T (WMMA f32) and Esum += Z, one pass per side ---
        const int wavesPerBlock = 4;                      // 128 threads = 4 waves
        int tilesU = (U + 15) / 16, tilesI = (I + 15) / 16;
        lowrank_wmma_fused<<<(tilesU + wavesPerBlock - 1) / wavesPerBlock,
                             wavesPerBlock * 32, 0, stream>>>(
            u_mul_s, T_iu, Znxt_u, Eu_sum, Gu_sum, U, Q);
        lowrank_wmma_fused<<<(tilesI + wavesPerBlock - 1) / wavesPerBlock,
                             wavesPerBlock * 32, 0, stream>>>(
            v_mul_s, T_ui, Znxt_i, Ei_sum, Gi_sum, I, Q);

        // double-buffer swap (host-side pointer swap; next layer re-zeroes Znxt)
        float* t;
        t = Ecur_u; Ecur_u = Znxt_u; Znxt_u = t;
        t = Ecur_i; Ecur_i = Znxt_i; Znxt_i = t;
    }
}